// redmarkjpeg_57543971831887
// MI455X (gfx1250) — compile-verified
//
#include <hip/hip_runtime.h>
#include <stdint.h>

typedef float v4f __attribute__((ext_vector_type(4)));
typedef int   v4i __attribute__((ext_vector_type(4)));
typedef __attribute__((address_space(1))) v4i g_v4i;   // global int4
typedef __attribute__((address_space(3))) v4i l_v4i;   // LDS int4

#define TPB      256
#define PLANE_F4 4096              // (128*128) floats per (n,c) plane, as float4
#define ITERS    (PLANE_F4 / TPB)  // 16 pipeline iterations per block

#if defined(__has_builtin)
#  if __has_builtin(__builtin_amdgcn_global_load_async_to_lds_b128)
#    define USE_ASYNC 1
#  else
#    define USE_ASYNC 0
#  endif
#else
#  define USE_ASYNC 0
#endif

// y = (x * (255/q) + noise) * (q/255), q broadcast per channel.
// Pure streaming kernel: 12 B traffic / 3 flops per element -> HBM bound
// (~17 us floor at 23.3 TB/s). Data path: CDNA5 async global->LDS engine,
// double-buffered, per-lane self-copy (no barriers; per-wave ASYNCcnt only),
// non-temporal b128 stores so the 128 MB output doesn't churn L2.
__global__ __launch_bounds__(TPB)
void jpeg_qnoise_kernel(const float* __restrict__ xg,
                        const float* __restrict__ qg,
                        const float* __restrict__ ng,
                        float* __restrict__ yg,
                        int C)
{
    const int tid   = threadIdx.x;
    const int plane = blockIdx.x;            // plane = n*C + c

    // Block-uniform quantization scalar: one s_load + uniform math per block.
    const float q = qg[plane % C];
    const float s = 255.0f / q;
    const float t = q / 255.0f;

    const v4f* __restrict__ xin = (const v4f*)xg + (size_t)plane * PLANE_F4;
    const v4f* __restrict__ nin = (const v4f*)ng + (size_t)plane * PLANE_F4;
    v4f* __restrict__ yout      = (v4f*)yg       + (size_t)plane * PLANE_F4;

#if USE_ASYNC
    __shared__ v4f lx[2][TPB];   // 4 KB per stage per tensor -> 16 KB LDS total
    __shared__ v4f ln[2][TPB];

    // Each lane async-copies its own 16 B into its own LDS slot and later
    // reads back the same slot: ASYNCcnt is per-wave, so s_wait_asynccnt is
    // the only synchronization required.
#define ISSUE_STAGE(i)                                          \
    do {                                                        \
        const int _st = (i) & 1;                                \
        __builtin_amdgcn_global_load_async_to_lds_b128(         \
            (g_v4i*)(xin + (i) * TPB + tid),                    \
            (l_v4i*)&lx[_st][tid], 0, 0);                       \
        __builtin_amdgcn_global_load_async_to_lds_b128(         \
            (g_v4i*)(nin + (i) * TPB + tid),                    \
            (l_v4i*)&ln[_st][tid], 0, 0);                       \
    } while (0)

    ISSUE_STAGE(0);
    ISSUE_STAGE(1);

#pragma unroll
    for (int i = 0; i < ITERS; ++i) {
        // 2 stages in flight = 4 outstanding async ops; waiting to <=2 means
        // stage i has landed in LDS while stage i+1 is still streaming.
        if (i + 1 < ITERS)
            asm volatile("s_wait_asynccnt 0x2" ::: "memory");
        else
            asm volatile("s_wait_asynccnt 0x0" ::: "memory");

        v4f xv = lx[i & 1][tid];   // ds_load_b128
        v4f nv = ln[i & 1][tid];   // ds_load_b128
        v4f r  = (xv * s + nv) * t;
        __builtin_nontemporal_store(r, yout + i * TPB + tid);

        if (i + 2 < ITERS)
            ISSUE_STAGE(i + 2);    // overwrites buffer (i&1): safe, the ds read
                                   // above retires long before the HBM round trip
    }
#undef ISSUE_STAGE
#else
    // Fallback: plain b128 streaming loads (toolchain lacks the async builtin).
#pragma unroll 4
    for (int i = tid; i < PLANE_F4; i += TPB) {
        v4f xv = xin[i];
        v4f nv = nin[i];
        v4f r  = (xv * s + nv) * t;
        __builtin_nontemporal_store(r, yout + i);
    }
#endif
}

extern "C" void kernel_launch(void* const* d_in, const int* in_sizes, int n_in,
                              void* d_out, int out_size, void* d_ws, size_t ws_size,
                              hipStream_t stream)
{
    const float* x = (const float*)d_in[0];   // (N,C,H,W) fp32
    const float* q = (const float*)d_in[1];   // (C,) fp32 quant table
    const float* n = (const float*)d_in[2];   // (N,C,H,W) fp32
    float* y = (float*)d_out;

    const int C      = in_sizes[1];           // 64
    const int planes = in_sizes[0] >> 14;     // total / (128*128) = N*C = 2048

    jpeg_qnoise_kernel<<<planes, TPB, 0, stream>>>(x, q, n, y, C);

    (void)n_in; (void)out_size; (void)d_ws; (void)ws_size;
}